// VGAE_44220983280304
// MI455X (gfx1250) — compile-verified
//
#include <hip/hip_runtime.h>

typedef __attribute__((ext_vector_type(2))) float v2f;
typedef __attribute__((ext_vector_type(8))) float v8f;

#define N_NODES 50000
#define D_IN    128
#define D_H     128
#define D_Z     64
#define N_EDGES 800000
#define N_EDGES_SL (N_EDGES + N_NODES)   // with self-loops
#define M_TILES (N_NODES / 16)           // 3125

// ---------------------------------------------------------------------------
// Utility fill
// ---------------------------------------------------------------------------
__global__ void fill_f32(float* p, float v, int n) {
  int i = blockIdx.x * blockDim.x + threadIdx.x;
  if (i < n) p[i] = v;
}

// ---------------------------------------------------------------------------
// Degree accumulation (self-loop contribution pre-initialized to 1.0)
// ---------------------------------------------------------------------------
__global__ void deg_scatter(const long long* __restrict__ dst, float* __restrict__ deg, int e) {
  int i = blockIdx.x * blockDim.x + threadIdx.x;
  if (i < e) unsafeAtomicAdd(&deg[dst[i]], 1.0f);
}

__global__ void deg_rsqrt(float* __restrict__ dis, int n) {
  int i = blockIdx.x * blockDim.x + threadIdx.x;
  if (i < n) dis[i] = rsqrtf(dis[i]);   // deg >= 1 always (self loops)
}

// ---------------------------------------------------------------------------
// WMMA fp32 GEMM, weights staged in LDS (pair-interleaved for ds_load_b64):
//   C[M,NN] = A[M,KK] @ B[KK,NN]
// One wave computes a 16 x NN output panel (NN/16 accumulators).
// LDS layout: Bs[p][c][j] = B[2p+j][c], linear index p*2*NN + 2*c + j.
// Lane reads (B[ka][n], B[ka+1][n]) as one contiguous 8-byte LDS load
// (ka = k + 2*half is always even since k steps by 4).
// ---------------------------------------------------------------------------
template <int NN, int KK>
__global__ __launch_bounds__(256) void gemm_wmma_lds(const float* __restrict__ A,
                                                     const float* __restrict__ B,
                                                     float* __restrict__ C,
                                                     int mtiles) {
  extern __shared__ float Bs[];  // KK*NN floats, pair-interleaved
  const int tid = threadIdx.x;

  // cooperative staged load + swizzle of the whole weight matrix
  for (int idx = tid; idx < (KK / 2) * NN; idx += 256) {
    const int p = idx / NN;  // pair row (rows 2p, 2p+1)
    const int c = idx % NN;
    v2f pr;
    pr.x = B[(size_t)(2 * p) * NN + c];
    pr.y = B[(size_t)(2 * p + 1) * NN + c];
    *(v2f*)&Bs[p * 2 * NN + 2 * c] = pr;
  }
  __syncthreads();

  const int wave = tid >> 5;
  const int lane = tid & 31;
  const int half = lane >> 4;   // 0: lanes 0-15, 1: lanes 16-31
  const int mn   = lane & 15;   // A row within tile / B,C column within tile

  const int tile = blockIdx.x * 8 + wave;
  if (tile >= mtiles) return;   // uniform per wave: skips WMMA entirely

  const float* Arow = A + (size_t)(tile * 16 + mn) * KK;

  constexpr int NT = NN / 16;
  v8f acc[NT] = {};

  for (int k = 0; k < KK; k += 4) {
    v2f a = *(const v2f*)(Arow + k + 2 * half);               // global_load_b64
    const float* brow = &Bs[(k / 2 + half) * (2 * NN) + 2 * mn];
#pragma unroll
    for (int nt = 0; nt < NT; ++nt) {
      v2f b = *(const v2f*)(brow + 32 * nt);                  // ds_load_b64
      acc[nt] = __builtin_amdgcn_wmma_f32_16x16x4_f32(
          false, a, false, b, (short)0, acc[nt], false, false);
    }
  }

  float* Cbase = C + (size_t)(tile * 16) * NN + mn;
#pragma unroll
  for (int nt = 0; nt < NT; ++nt) {
#pragma unroll
    for (int r = 0; r < 8; ++r) {
      Cbase[(size_t)(r + 8 * half) * NN + nt * 16] = acc[nt][r];
    }
  }
}

// ---------------------------------------------------------------------------
// Dual-weight variant: C0 = A @ B0, C1 = A @ B1 (B0,B1: [KK,64]) sharing the
// A stream. Combined LDS panel is 128 columns wide (B0 cols 0-63, B1 64-127).
// ---------------------------------------------------------------------------
__global__ __launch_bounds__(256) void gemm_wmma_dual_lds(const float* __restrict__ A,
                                                          const float* __restrict__ B0,
                                                          const float* __restrict__ B1,
                                                          float* __restrict__ C0,
                                                          float* __restrict__ C1,
                                                          int mtiles) {
  extern __shared__ float Bs[];  // 128*128 floats pair-interleaved (KK=128)
  const int tid = threadIdx.x;

  for (int idx = tid; idx < 64 * 128; idx += 256) {
    const int p = idx >> 7;      // pair row
    const int c = idx & 127;     // combined column
    const float* Bsel = (c < D_Z) ? B0 : B1;
    const int cc = c & (D_Z - 1);
    v2f pr;
    pr.x = Bsel[(size_t)(2 * p) * D_Z + cc];
    pr.y = Bsel[(size_t)(2 * p + 1) * D_Z + cc];
    *(v2f*)&Bs[p * 256 + 2 * c] = pr;
  }
  __syncthreads();

  const int wave = tid >> 5;
  const int lane = tid & 31;
  const int half = lane >> 4;
  const int mn   = lane & 15;

  const int tile = blockIdx.x * 8 + wave;
  if (tile >= mtiles) return;

  const float* Arow = A + (size_t)(tile * 16 + mn) * D_H;

  v8f acc[8] = {};
  for (int k = 0; k < D_H; k += 4) {
    v2f a = *(const v2f*)(Arow + k + 2 * half);
    const float* brow = &Bs[(k / 2 + half) * 256 + 2 * mn];
#pragma unroll
    for (int nt = 0; nt < 8; ++nt) {
      v2f b = *(const v2f*)(brow + 32 * nt);
      acc[nt] = __builtin_amdgcn_wmma_f32_16x16x4_f32(
          false, a, false, b, (short)0, acc[nt], false, false);
    }
  }

  float* C0base = C0 + (size_t)(tile * 16) * D_Z + mn;
  float* C1base = C1 + (size_t)(tile * 16) * D_Z + mn;
#pragma unroll
  for (int nt = 0; nt < 8; ++nt) {
    float* Cb = (nt < 4) ? C0base : C1base;
    const int col = (nt & 3) * 16;
#pragma unroll
    for (int r = 0; r < 8; ++r) {
      Cb[(size_t)(r + 8 * half) * D_Z + col] = acc[nt][r];
    }
  }
}

// ---------------------------------------------------------------------------
// Edge scatter conv1: one wave per edge, float4 per lane (128 features).
//   h[dst] += xw[src] * dis[src]*dis[dst]
// ---------------------------------------------------------------------------
__global__ void scatter_h(const long long* __restrict__ src,
                          const long long* __restrict__ dst,
                          const float* __restrict__ dis,
                          const float* __restrict__ xw,
                          float* __restrict__ h) {
  const int gid = blockIdx.x * blockDim.x + threadIdx.x;
  const int e = gid >> 5;       // wave index == edge index
  const int lane = gid & 31;
  if (e >= N_EDGES_SL) return;
  long long s, d;
  if (e < N_EDGES) { s = src[e]; d = dst[e]; }
  else             { s = d = (long long)(e - N_EDGES); }
  const float norm = dis[s] * dis[d];
  const float4 v = *(const float4*)(xw + s * D_H + lane * 4);  // global_load_b128
  float* hp = h + d * D_H + lane * 4;
  unsafeAtomicAdd(hp + 0, v.x * norm);
  unsafeAtomicAdd(hp + 1, v.y * norm);
  unsafeAtomicAdd(hp + 2, v.z * norm);
  unsafeAtomicAdd(hp + 3, v.w * norm);
}

__global__ void bias_relu(float* __restrict__ h, const float* __restrict__ b, int n) {
  int i = blockIdx.x * blockDim.x + threadIdx.x;
  if (i < n) {
    float v = h[i] + b[i & (D_H - 1)];
    h[i] = v > 0.0f ? v : 0.0f;
  }
}

// ---------------------------------------------------------------------------
// Edge scatter conv2/3: one wave per edge; lanes 0-15 handle mu (64 feats as
// float4), lanes 16-31 handle logstd.
// ---------------------------------------------------------------------------
__global__ void scatter_muls(const long long* __restrict__ src,
                             const long long* __restrict__ dst,
                             const float* __restrict__ dis,
                             const float* __restrict__ hmu,
                             const float* __restrict__ hls,
                             float* __restrict__ omu,
                             float* __restrict__ ols) {
  const int gid = blockIdx.x * blockDim.x + threadIdx.x;
  const int e = gid >> 5;
  const int lane = gid & 31;
  if (e >= N_EDGES_SL) return;
  long long s, d;
  if (e < N_EDGES) { s = src[e]; d = dst[e]; }
  else             { s = d = (long long)(e - N_EDGES); }
  const float norm = dis[s] * dis[d];
  const int q = lane >> 4;          // 0: mu, 1: logstd
  const int f4 = (lane & 15) * 4;
  const float* sb = q ? hls : hmu;
  float*       db = q ? ols : omu;
  const float4 v = *(const float4*)(sb + s * D_Z + f4);
  float* op = db + d * D_Z + f4;
  unsafeAtomicAdd(op + 0, v.x * norm);
  unsafeAtomicAdd(op + 1, v.y * norm);
  unsafeAtomicAdd(op + 2, v.z * norm);
  unsafeAtomicAdd(op + 3, v.w * norm);
}

// ---------------------------------------------------------------------------
// Finalize: mu += bmu; ls += bls; z = mu + eps*exp(ls)
// ---------------------------------------------------------------------------
__global__ void finalize(float* __restrict__ z,
                         float* __restrict__ mu,
                         float* __restrict__ ls,
                         const float* __restrict__ bmu,
                         const float* __restrict__ bls,
                         const float* __restrict__ eps, int n) {
  int i = blockIdx.x * blockDim.x + threadIdx.x;
  if (i < n) {
    const int f = i & (D_Z - 1);
    const float m = mu[i] + bmu[f];
    const float l = ls[i] + bls[f];
    mu[i] = m;
    ls[i] = l;
    z[i]  = m + eps[i] * expf(l);
  }
}

// ---------------------------------------------------------------------------
extern "C" void kernel_launch(void* const* d_in, const int* in_sizes, int n_in,
                              void* d_out, int out_size, void* d_ws, size_t ws_size,
                              hipStream_t stream) {
  const float*     x    = (const float*)d_in[0];
  const long long* eidx = (const long long*)d_in[1];   // [2, E] int64
  const float*     eps  = (const float*)d_in[2];
  const float*     W1   = (const float*)d_in[3];
  const float*     b1   = (const float*)d_in[4];
  const float*     Wmu  = (const float*)d_in[5];
  const float*     bmu  = (const float*)d_in[6];
  const float*     Wls  = (const float*)d_in[7];
  const float*     bls  = (const float*)d_in[8];

  const long long* src = eidx;            // row 0
  const long long* dst = eidx + N_EDGES;  // row 1

  float* out = (float*)d_out;
  float* z_out  = out;                           // [N, 64]
  float* mu_out = out + (size_t)N_NODES * D_Z;   // [N, 64] (accumulator, then final)
  float* ls_out = out + (size_t)2 * N_NODES * D_Z;

  // workspace layout (floats); all sub-buffers stay 16B-aligned
  float* ws  = (float*)d_ws;
  float* dis = ws;                                   // [N]
  float* xw  = dis + N_NODES;                        // [N, 128]  (reused: hmu/hls)
  float* h   = xw + (size_t)N_NODES * D_H;           // [N, 128]
  float* hmu = xw;                                   // [N, 64] after xw is dead
  float* hls = xw + (size_t)N_NODES * D_Z;           // [N, 64]

  const int NT = 256;
  const int n_h   = N_NODES * D_H;   // 6.4M
  const int n_z   = N_NODES * D_Z;   // 3.2M
  const int n_out = 3 * n_z;         // 9.6M

  // 1) init: deg=1 (self-loop), hidden accumulator=0, d_out=0 (mu/ls accumulate there)
  fill_f32<<<(N_NODES + NT - 1) / NT, NT, 0, stream>>>(dis, 1.0f, N_NODES);
  fill_f32<<<(n_h + NT - 1) / NT, NT, 0, stream>>>(h, 0.0f, n_h);
  fill_f32<<<(n_out + NT - 1) / NT, NT, 0, stream>>>(out, 0.0f, n_out);

  // 2) degree + inverse sqrt
  deg_scatter<<<(N_EDGES + NT - 1) / NT, NT, 0, stream>>>(dst, dis, N_EDGES);
  deg_rsqrt<<<(N_NODES + NT - 1) / NT, NT, 0, stream>>>(dis, N_NODES);

  // 3) xw = x @ W1   (16-row panel per wave, W1 staged in 64KB LDS)
  {
    const int grid = (M_TILES + 7) / 8;                  // 391
    const size_t shmem = (size_t)D_IN * D_H * sizeof(float);  // 64KB
    gemm_wmma_lds<D_H, D_IN><<<grid, 256, shmem, stream>>>(x, W1, xw, M_TILES);
  }

  // 4) aggregate + bias + relu -> hidden (wave per edge, fp32 atomics)
  scatter_h<<<(N_EDGES_SL * 32) / NT, NT, 0, stream>>>(src, dst, dis, xw, h);
  bias_relu<<<(n_h + NT - 1) / NT, NT, 0, stream>>>(h, b1, n_h);

  // 5) hmu = h @ Wmu, hls = h @ Wls fused (shared A stream, Wmu|Wls in LDS)
  {
    const int grid = (M_TILES + 7) / 8;                  // 391
    const size_t shmem = (size_t)D_H * 2 * D_Z * sizeof(float);  // 64KB
    gemm_wmma_dual_lds<<<grid, 256, shmem, stream>>>(h, Wmu, Wls, hmu, hls, M_TILES);
  }

  // 6) aggregate mu and logstd directly into d_out regions
  scatter_muls<<<(N_EDGES_SL * 32) / NT, NT, 0, stream>>>(src, dst, dis, hmu, hls,
                                                          mu_out, ls_out);

  // 7) biases + reparameterize
  finalize<<<(n_z + NT - 1) / NT, NT, 0, stream>>>(z_out, mu_out, ls_out, bmu, bls, eps, n_z);
}